// ATSSAssigner_3633542333201
// MI455X (gfx1250) — compile-verified
//
#include <hip/hip_runtime.h>
#include <hip/hip_bf16.h>

// ---------------------------------------------------------------------------
// ATSS assigner for MI455X (gfx1250, wave32).
// Distance matrix computed on the matrix pipe: pairwise squared distance as a
// rank-4 matmul via V_WMMA_F32_16X16X4_F32. Intermediate (34 MB) stays in the
// 192 MB L2. Score output (43 MB) written with coalesced 32-lane strided
// stores; sparse phases are one-wave-per-work-item with shuffle reductions.
// ---------------------------------------------------------------------------

#define NA        8400
#define NMAX      64
#define NCLS      80
#define NCAND     27
#define FLT_BIG   __FLT_MAX__
#define INT_BIG   0x7FFFFFFF
#define TILES_PER_WAVE 15   // 240 anchors per wave; 8400/240 = 35

typedef __attribute__((ext_vector_type(2))) float v2f;
typedef __attribute__((ext_vector_type(8))) float v8f;

__device__ __forceinline__ bool lexLess(float d0, int i0, float d1, int i1) {
    return d0 < d1 || (d0 == d1 && i0 < i1);
}

// ---------------- K0: init scratch ----------------
__global__ void k_init(int* __restrict__ pos_cnt, int* __restrict__ pos_g, int n) {
    int i = blockIdx.x * blockDim.x + threadIdx.x;
    if (i < n) { pos_cnt[i] = 0; pos_g[i] = -1; }
}

// ---------------- K1: pairwise squared distances via WMMA ----------------
// dist2[(b*64+g)][a] = |gt_center - anc_center|^2 as a K=4 matmul:
//   A row g = [gx, gy, gx^2+gy^2, 1]        (16x4, f32)
//   B col a = [-2ax, -2ay, 1, ax^2+ay^2]    (4x16, f32)
// ISA 7.12.2 layouts: A: lanes 0-15 hold K0/K1, lanes 16-31 hold K2/K3 (M=lane&15).
//                     B: lanes 0-15 hold K0/K1, lanes 16-31 hold K2/K3 (N=lane&15).
//                     C: VGPR r, lanes 0-15 -> (M=r, N=lane); lanes 16-31 -> (M=r+8).
__global__ void __launch_bounds__(32)
k_dist_wmma(const float* __restrict__ anc, const float* __restrict__ gt,
            float* __restrict__ dist)
{
    const int lane = threadIdx.x;                 // 0..31, one wave per block
    const int bt   = blockIdx.x;                  // gt rows bt*16 .. bt*16+15
    const int n0   = blockIdx.y * (16 * TILES_PER_WAVE);

    const int row  = lane & 15;
    const int grow = bt * 16 + row;               // flat gt row = b*64 + g
    const float4 gb = ((const float4*)gt)[grow];
    const float gx = (gb.x + gb.z) * 0.5f;
    const float gy = (gb.y + gb.w) * 0.5f;

    v2f A;
    if (lane < 16) { A[0] = gx;              A[1] = gy;   }
    else           { A[0] = gx*gx + gy*gy;   A[1] = 1.0f; }

    for (int t = 0; t < TILES_PER_WAVE; ++t) {
        const int acol = n0 + t * 16 + (lane & 15);
        const float4 ab = ((const float4*)anc)[acol];
        const float ax = (ab.x + ab.z) * 0.5f;
        const float ay = (ab.y + ab.w) * 0.5f;
        v2f B;
        if (lane < 16) { B[0] = -2.0f * ax; B[1] = -2.0f * ay;     }
        else           { B[0] = 1.0f;       B[1] = ax*ax + ay*ay;  }

        v8f c = {};
        c = __builtin_amdgcn_wmma_f32_16x16x4_f32(false, A, false, B,
                                                  (short)0, c, false, false);

        const int colBase = n0 + t * 16 + (lane & 15);
        const int rhalf   = (lane >> 4) << 3;     // rows r or r+8 per lane half
        #pragma unroll
        for (int r = 0; r < 8; ++r)
            dist[(size_t)(bt * 16 + r + rhalf) * NA + colBase] = c[r];
    }
}

// ---------------- K2: top-9 per level, threshold, scatter positives ----------------
// One wave per (b,g). Distances read coalesced (lane-strided). Per-lane top-9
// kept unsorted in registers with lazy worst tracking; 9 rounds of wave-wide
// lexicographic min extraction via shuffles. Then candidate IoU stats
// (mean + std ddof=1), strict-inside test, and atomic scatter per anchor.
__global__ void __launch_bounds__(32)
k_select(const float* __restrict__ anc, const float* __restrict__ gt,
         const float* __restrict__ mask_gt, const float* __restrict__ dist,
         int* __restrict__ pos_cnt, int* __restrict__ pos_g)
{
    const int lane = threadIdx.x;
    const int bg   = blockIdx.x;                  // 0 .. bs*64-1
    const int b    = bg >> 6;
    const int g    = bg & 63;

    if (mask_gt[bg] <= 0.0f) return;              // masked gt contributes nothing

    const float4 gb = ((const float4*)gt)[bg];
    const float garea = (gb.z - gb.x) * (gb.w - gb.y);
    const float* drow = dist + (size_t)bg * NA;

    __shared__ int candLds[32];

    const int lvl_start[3] = {0, 6400, 8000};
    const int lvl_n[3]     = {6400, 1600, 400};

    for (int lv = 0; lv < 3; ++lv) {
        float bd[9]; int bi[9];
        #pragma unroll
        for (int k = 0; k < 9; ++k) { bd[k] = FLT_BIG; bi[k] = INT_BIG; }
        float wd = FLT_BIG; int wi = INT_BIG;     // current worst of the 9

        const int st = lvl_start[lv], nl = lvl_n[lv];
        for (int j = lane; j < nl; j += 32) {
            const int   idx = st + j;
            const float d2  = drow[idx];
            if (lexLess(d2, idx, wd, wi)) {
                bool replaced = false;
                float nwd = -FLT_BIG; int nwi = -1;
                #pragma unroll
                for (int k = 0; k < 9; ++k) {
                    if (!replaced && bd[k] == wd && bi[k] == wi) {
                        bd[k] = d2; bi[k] = idx; replaced = true;
                    }
                    if (bd[k] > nwd || (bd[k] == nwd && bi[k] > nwi)) {
                        nwd = bd[k]; nwi = bi[k];
                    }
                }
                wd = nwd; wi = nwi;
            }
        }

        // 9 rounds of wave-wide min extraction -> the 9 globally smallest
        for (int r = 0; r < 9; ++r) {
            float md = FLT_BIG; int mi = INT_BIG;
            #pragma unroll
            for (int k = 0; k < 9; ++k)
                if (lexLess(bd[k], bi[k], md, mi)) { md = bd[k]; mi = bi[k]; }
            #pragma unroll
            for (int off = 16; off > 0; off >>= 1) {
                const float od = __shfl_xor(md, off, 32);
                const int   oi = __shfl_xor(mi, off, 32);
                if (lexLess(od, oi, md, mi)) { md = od; mi = oi; }
            }
            #pragma unroll
            for (int k = 0; k < 9; ++k)          // winner retires its entry
                if (bd[k] == md && bi[k] == mi) { bd[k] = FLT_BIG; bi[k] = INT_BIG; }
            if (lane == 0) candLds[lv * 9 + r] = mi;
        }
    }
    __syncthreads();

    // Candidate IoU (_pairwise_iou: raw areas, union clamped at 1e-6) + inside test
    float iou = 0.0f; bool ing = false; int aidx = -1;
    if (lane < NCAND) {
        aidx = candLds[lane];
        const float4 ab = ((const float4*)anc)[aidx];
        const float a2  = (ab.z - ab.x) * (ab.w - ab.y);
        const float w0  = fmaxf(fminf(gb.z, ab.z) - fmaxf(gb.x, ab.x), 0.0f);
        const float h0  = fmaxf(fminf(gb.w, ab.w) - fmaxf(gb.y, ab.y), 0.0f);
        const float ov  = w0 * h0;
        iou = ov / fmaxf(garea + a2 - ov, 1e-6f);
        const float acx = (ab.x + ab.z) * 0.5f, acy = (ab.y + ab.w) * 0.5f;
        const float m = fminf(fminf(acx - gb.x, acy - gb.y),
                              fminf(gb.z - acx, gb.w - acy));
        ing = m > 1e-9f;
    }
    float s = iou, s2 = iou * iou;
    #pragma unroll
    for (int off = 16; off > 0; off >>= 1) {
        s  += __shfl_xor(s,  off, 32);
        s2 += __shfl_xor(s2, off, 32);
    }
    const float mean = s / 27.0f;
    const float var  = fmaxf(s2 - 27.0f * mean * mean, 0.0f) / 26.0f;
    const float thr  = mean + sqrtf(var);

    if (lane < NCAND && iou > thr && ing) {
        atomicAdd(&pos_cnt[b * NA + aidx], 1);
        atomicMax(&pos_g  [b * NA + aidx], g);
    }
}

// ---------------- K3: finalize per anchor ----------------
// One wave per (b,a). Conflicts (cnt>1) resolved by argmax IoU over all 64 gts
// (2 per lane + shuffle argmax, low-index tie break, matching jnp.argmax).
__global__ void __launch_bounds__(256)
k_final(const float* __restrict__ anc, const int* __restrict__ gt_labels,
        const float* __restrict__ gt, const float* __restrict__ pd,
        const int* __restrict__ pos_cnt, const int* __restrict__ pos_g,
        float* __restrict__ out_labels, float* __restrict__ out_bboxes,
        float* __restrict__ out_scores, float* __restrict__ out_fg, int total)
{
    const int lane = threadIdx.x & 31;
    const int w = blockIdx.x * (blockDim.x >> 5) + (threadIdx.x >> 5);
    if (w >= total) return;
    const int b = w / NA;
    const int a = w - b * NA;

    const int cnt = pos_cnt[w];
    int   gstar = 0;
    int   label = NCLS;
    float piou  = 0.0f;
    float4 tb;

    if (cnt == 1) {
        gstar = pos_g[w];
    } else if (cnt > 1) {
        const float4 ab = ((const float4*)anc)[a];
        const float aarea = (ab.z - ab.x) * (ab.w - ab.y);
        float bestI = -FLT_BIG; int bestG = INT_BIG;
        #pragma unroll
        for (int rep = 0; rep < 2; ++rep) {
            const int gg = lane + rep * 32;
            const float4 q  = ((const float4*)gt)[b * NMAX + gg];
            const float qa  = (q.z - q.x) * (q.w - q.y);
            const float w0  = fmaxf(fminf(q.z, ab.z) - fmaxf(q.x, ab.x), 0.0f);
            const float h0  = fmaxf(fminf(q.w, ab.w) - fmaxf(q.y, ab.y), 0.0f);
            const float ov  = w0 * h0;
            const float i0  = ov / fmaxf(qa + aarea - ov, 1e-6f);
            if (i0 > bestI || (i0 == bestI && gg < bestG)) { bestI = i0; bestG = gg; }
        }
        #pragma unroll
        for (int off = 16; off > 0; off >>= 1) {
            const float oi = __shfl_xor(bestI, off, 32);
            const int   og = __shfl_xor(bestG, off, 32);
            if (oi > bestI || (oi == bestI && og < bestG)) { bestI = oi; bestG = og; }
        }
        gstar = bestG;
    }

    if (cnt > 0) {
        tb    = ((const float4*)gt)[b * NMAX + gstar];
        label = gt_labels[b * NMAX + gstar];
        // _batch_iou: clamped areas, +1e-9 in denominator
        const float4 p  = ((const float4*)pd)[w];
        const float ov  = fmaxf(fminf(tb.z, p.z) - fmaxf(tb.x, p.x), 0.0f)
                        * fmaxf(fminf(tb.w, p.w) - fmaxf(tb.y, p.y), 0.0f);
        const float a1  = fmaxf(tb.z - tb.x, 0.0f) * fmaxf(tb.w - tb.y, 0.0f);
        const float a2  = fmaxf(p.z - p.x, 0.0f) * fmaxf(p.w - p.y, 0.0f);
        piou = ov / (a1 + a2 - ov + 1e-9f);
    } else {
        tb = ((const float4*)gt)[b * NMAX];   // argmax of all-zero column -> gt 0
    }

    if (lane == 0) {
        out_labels[w] = (float)label;
        out_fg[w]     = (cnt > 0) ? 1.0f : 0.0f;
    }
    if (lane < 4) {
        const float v = (lane == 0) ? tb.x : (lane == 1) ? tb.y
                      : (lane == 2) ? tb.z : tb.w;
        out_bboxes[(size_t)w * 4 + lane] = v;
    }
    float* sc = out_scores + (size_t)w * NCLS;
    for (int i = lane; i < NCLS; i += 32)
        sc[i] = (i == label && cnt > 0) ? piou : 0.0f;
}

// ---------------------------------------------------------------------------
extern "C" void kernel_launch(void* const* d_in, const int* in_sizes, int n_in,
                              void* d_out, int out_size, void* d_ws, size_t ws_size,
                              hipStream_t stream) {
    const float* anc = (const float*)d_in[0];   // (8400,4)
    const int*   gtl = (const int*  )d_in[1];   // (bs,64,1)
    const float* gtb = (const float*)d_in[2];   // (bs,64,4)
    const float* mgt = (const float*)d_in[3];   // (bs,64,1)
    const float* pdb = (const float*)d_in[4];   // (bs,8400,4)

    const int bs    = in_sizes[1] / NMAX;
    const int total = bs * NA;

    // Workspace: dist2 (bs*64 x 8400 f32) | pos_cnt | pos_g
    char*  ws    = (char*)d_ws;
    float* dist  = (float*)ws;
    const size_t distBytes = (size_t)bs * NMAX * NA * sizeof(float);
    int* pos_cnt = (int*)(ws + distBytes);
    int* pos_g   = pos_cnt + total;

    // Outputs, concatenated flat in return order (all as f32)
    float* out_labels = (float*)d_out;
    float* out_bboxes = out_labels + total;
    float* out_scores = out_bboxes + (size_t)total * 4;
    float* out_fg     = out_scores + (size_t)total * NCLS;

    k_init<<<(total + 255) / 256, 256, 0, stream>>>(pos_cnt, pos_g, total);

    dim3 g1(bs * 4, NA / (16 * TILES_PER_WAVE));      // (bs*4, 35)
    k_dist_wmma<<<g1, 32, 0, stream>>>(anc, gtb, dist);

    k_select<<<bs * NMAX, 32, 0, stream>>>(anc, gtb, mgt, dist, pos_cnt, pos_g);

    k_final<<<(total + 7) / 8, 256, 0, stream>>>(anc, gtl, gtb, pdb,
                                                 pos_cnt, pos_g,
                                                 out_labels, out_bboxes,
                                                 out_scores, out_fg, total);
}